// MeshCNNBlock_54262616818057
// MI455X (gfx1250) — compile-verified
//
#include <hip/hip_runtime.h>

typedef float v2f __attribute__((ext_vector_type(2)));
typedef float v8f __attribute__((ext_vector_type(8)));

#define B_        8
#define CIN       16
#define COUT      32
#define E_        120000
#define TILES_PB  7500          // E_/16
#define TOTAL_T   60000         // B_*TILES_PB
#define GROUPS    15000         // TOTAL_T / 4 waves
#define NBLK      1200
#define NELEM     30720000      // B_*COUT*E_
#define NPOS_F    960000.0f     // B_*E_

// f_lds element layout: pair-interleaved so each lane's A fragment (K,K+1)
// for a row is one aligned 8-byte DS load:
//   elem(K,row) = (K>>1)*32 + row*2 + (K&1)
__device__ __forceinline__ int fidx(int K, int row) {
    return ((K >> 1) << 5) + (row << 1) + (K & 1);
}

// ---------------------------------------------------------------------------
// Kernel 1: mesh-conv as WMMA GEMM (M=16-edge tile, K=80, N=32) + BN partials
// ---------------------------------------------------------------------------
__global__ __launch_bounds__(128)
void meshconv_wmma_kernel(const float* __restrict__ x,
                          const int*   __restrict__ gidx,
                          const float* __restrict__ W,
                          float*       __restrict__ y,
                          float*       __restrict__ part_sum,
                          float*       __restrict__ part_sq)
{
    // B-fragments in exact WMMA layout: [ntile][kk][lane] -> (K0,K0+1) pair
    __shared__ v2f   wb[2][20][32];          // 10 KB
    __shared__ float f_lds[4][80 * 16];      // 20 KB  (per-wave A staging, pair-interleaved)
    __shared__ int   idx_lds[4][64];         // 1 KB   (16 edges x 4 neighbor ids)
    __shared__ float bn_s[32][8];            // 1 KB
    __shared__ float bn_q[32][8];            // 1 KB

    const int tid  = threadIdx.x;
    const int wave = tid >> 5;
    const int lane = tid & 31;
    const int hrow = lane & 15;     // M row / N col within half-wave
    const int hsel = lane >> 4;     // 0: lanes 0-15, 1: lanes 16-31

    // Stage W[o][c][s] into B-fragment layout. K = c*5 + s.
    // VGPR0 (.x): lanes0-15 K=4kk, lanes16-31 K=4kk+2 ; VGPR1 (.y): K+1.
    for (int j = tid; j < 1280; j += 128) {
        int ntile = j / 640;
        int rem   = j % 640;
        int kk    = rem >> 5;
        int ln    = rem & 31;
        int K0    = 4 * kk + (ln >> 4) * 2;
        int o     = ntile * 16 + (ln & 15);
        v2f v;
        v.x = W[(o * CIN + (K0 / 5)) * 5 + (K0 % 5)];
        v.y = W[(o * CIN + ((K0 + 1) / 5)) * 5 + ((K0 + 1) % 5)];
        wb[ntile][kk][ln] = v;
    }
    __syncthreads();

    float ls0 = 0.f, lq0 = 0.f, ls1 = 0.f, lq1 = 0.f;

    for (int g = blockIdx.x; g < GROUPS; g += gridDim.x) {
        const int tile = g * 4 + wave;              // < TOTAL_T always
        const int b    = tile / TILES_PB;
        const int e0   = (tile % TILES_PB) * 16;

        // Stage the 64 neighbor indices of this 16-edge tile.
        const int* gp = gidx + (b * E_ + e0) * 4;
        idx_lds[wave][lane]      = gp[lane];
        idx_lds[wave][lane + 32] = gp[lane + 32];
        __syncthreads();

        // Build symmetric features. K = c*5 + s, pair-interleaved layout.
        const float* xb = x + b * CIN * E_;
        #pragma unroll
        for (int it = 0; it < 8; ++it) {
            const int c   = it * 2 + hsel;          // channel 0..15
            const int row = hrow;                   // edge-in-tile 0..15
            const float* xc = xb + c * E_;
            float ctr = xc[e0 + row];
            int i0 = idx_lds[wave][row * 4 + 0];
            int i1 = idx_lds[wave][row * 4 + 1];
            int i2 = idx_lds[wave][row * 4 + 2];
            int i3 = idx_lds[wave][row * 4 + 3];
            float na = xc[i0], nb = xc[i1], nc = xc[i2], nd = xc[i3];
            const int K = c * 5;
            float* fw = f_lds[wave];
            fw[fidx(K + 0, row)] = ctr;
            fw[fidx(K + 1, row)] = fabsf(na - nc);
            fw[fidx(K + 2, row)] = na + nc;
            fw[fidx(K + 3, row)] = fabsf(nb - nd);
            fw[fidx(K + 4, row)] = nb + nd;
        }
        __syncthreads();

        // 40 x V_WMMA_F32_16X16X4_F32 (fully unrolled K=80, N=32 in 2 tiles).
        // A fragment: lane<16 holds (K,K+1), lane>=16 holds (K+2,K+3) of its row
        //             = pair index (2*kk + hsel), one b64 DS load per step.
        const v2f* f2 = reinterpret_cast<const v2f*>(f_lds[wave]) + hrow;
        v8f acc0 = {}; v8f acc1 = {};
        #pragma unroll
        for (int kk = 0; kk < 20; ++kk) {
            v2f a = f2[(2 * kk + hsel) * 16];   // ((2kk+hsel)*32 + hrow*2) floats
            acc0 = __builtin_amdgcn_wmma_f32_16x16x4_f32(
                       false, a, false, wb[0][kk][lane], (short)0, acc0, false, false);
            acc1 = __builtin_amdgcn_wmma_f32_16x16x4_f32(
                       false, a, false, wb[1][kk][lane], (short)0, acc1, false, false);
        }

        // D layout: lane<16 VGPR j -> M=j ; lane>=16 -> M=j+8.  N = hrow (+16).
        const int o0 = hrow, o1 = 16 + hrow;
        float* y0 = y + (b * COUT + o0) * E_ + e0 + hsel * 8;
        float* y1 = y + (b * COUT + o1) * E_ + e0 + hsel * 8;
        #pragma unroll
        for (int i = 0; i < 8; ++i) {
            float v0 = acc0[i], v1 = acc1[i];
            y0[i] = v0;  ls0 += v0;  lq0 += v0 * v0;
            y1[i] = v1;  ls1 += v1;  lq1 += v1 * v1;
        }
    }

    // Deterministic per-block BN reduction: each (channel, slot) has one writer.
    const int slot = wave * 2 + hsel;
    bn_s[hrow][slot]      = ls0;  bn_q[hrow][slot]      = lq0;
    bn_s[16 + hrow][slot] = ls1;  bn_q[16 + hrow][slot] = lq1;
    __syncthreads();
    if (tid < 32) {
        float s = 0.f;
        #pragma unroll
        for (int j = 0; j < 8; ++j) s += bn_s[tid][j];
        part_sum[blockIdx.x * 32 + tid] = s;
    } else if (tid < 64) {
        const int c = tid - 32;
        float q = 0.f;
        #pragma unroll
        for (int j = 0; j < 8; ++j) q += bn_q[c][j];
        part_sq[blockIdx.x * 32 + c] = q;
    }
}

// ---------------------------------------------------------------------------
// Kernel 2: fold partials into per-channel scale/shift.
// Conv bias b cancels exactly under train-mode BN, so it is omitted entirely.
// ---------------------------------------------------------------------------
__global__ void bn_stats_kernel(const float* __restrict__ part_sum,
                                const float* __restrict__ part_sq,
                                const float* __restrict__ gamma,
                                const float* __restrict__ beta,
                                float*       __restrict__ ss)
{
    const int o = threadIdx.x;
    if (o >= 32) return;
    float s = 0.f, q = 0.f;
    for (int j = 0; j < NBLK; ++j) {
        s += part_sum[j * 32 + o];
        q += part_sq [j * 32 + o];
    }
    const float mean = s / NPOS_F;
    const float var  = q / NPOS_F - mean * mean;
    const float inv  = rsqrtf(var + 1e-5f);
    const float sc   = gamma[o] * inv;
    ss[o]      = sc;
    ss[32 + o] = beta[o] - mean * sc;
}

// ---------------------------------------------------------------------------
// Kernel 3: y = relu(y*scale + shift), float4-vectorized (E % 4 == 0).
// ---------------------------------------------------------------------------
__global__ __launch_bounds__(256)
void bn_relu_kernel(float* __restrict__ y, const float* __restrict__ ss)
{
    const int i4  = blockIdx.x * 256 + threadIdx.x;      // 7,680,000 float4 total
    const int o   = ((i4 * 4) / E_) & (COUT - 1);
    const float sc = ss[o], sh = ss[32 + o];
    float4 v = reinterpret_cast<float4*>(y)[i4];
    v.x = fmaxf(fmaf(v.x, sc, sh), 0.f);
    v.y = fmaxf(fmaf(v.y, sc, sh), 0.f);
    v.z = fmaxf(fmaf(v.z, sc, sh), 0.f);
    v.w = fmaxf(fmaf(v.w, sc, sh), 0.f);
    reinterpret_cast<float4*>(y)[i4] = v;
}

// ---------------------------------------------------------------------------
extern "C" void kernel_launch(void* const* d_in, const int* in_sizes, int n_in,
                              void* d_out, int out_size, void* d_ws, size_t ws_size,
                              hipStream_t stream)
{
    const float* x     = (const float*)d_in[0];   // [8,16,120000]
    const int*   gidx  = (const int*)  d_in[1];   // [8,120000,4]
    const float* W     = (const float*)d_in[2];   // [32,16,5]
    // d_in[3] = conv bias b : cancels in train-mode BN, unused.
    const float* gamma = (const float*)d_in[4];   // [32]
    const float* beta  = (const float*)d_in[5];   // [32]

    float* y  = (float*)d_out;                    // [8,32,120000]
    float* ws = (float*)d_ws;
    float* part_sum = ws;                         // NBLK*32
    float* part_sq  = ws + NBLK * 32;             // NBLK*32
    float* ss       = ws + 2 * NBLK * 32;         // 64 (scale | shift)

    meshconv_wmma_kernel<<<NBLK, 128, 0, stream>>>(x, gidx, W, y, part_sum, part_sq);
    bn_stats_kernel<<<1, 32, 0, stream>>>(part_sum, part_sq, gamma, beta, ss);
    bn_relu_kernel<<<NELEM / 4 / 256, 256, 0, stream>>>(y, ss);
}